// LinearAdderV2_30133490549220
// MI455X (gfx1250) — compile-verified
//
#include <hip/hip_runtime.h>

typedef float v2f __attribute__((ext_vector_type(2)));
typedef float v8f __attribute__((ext_vector_type(8)));

#define B_ROWS 8192
#define O_COLS 2048
#define IN_DIM 2048

// ---------------------------------------------------------------------------
// Pass 1: row-sums. sums[0..8191] = rowsum(x), sums[8192..10239] = rowsum(w).
// One wave (32 lanes) per row; each lane reads 16 float4 chunks (64 floats).
// Bandwidth-bound: 80 MiB of reads.
// ---------------------------------------------------------------------------
__global__ void __launch_bounds__(256)
rowsum_kernel(const float* __restrict__ x,
              const float* __restrict__ w,
              float* __restrict__ sums) {
    const int wave = threadIdx.x >> 5;              // 8 waves / block
    const int lane = threadIdx.x & 31;
    const int row  = blockIdx.x * 8 + wave;         // 0..10239

    const float* __restrict__ src =
        (row < B_ROWS) ? (x + (size_t)row * IN_DIM)
                       : (w + (size_t)(row - B_ROWS) * IN_DIM);

    float acc = 0.0f;
#pragma unroll
    for (int i = 0; i < IN_DIM / 128; ++i) {        // 16 iterations
        const float4 v = *(const float4*)(src + i * 128 + lane * 4);
        acc += (v.x + v.y) + (v.z + v.w);
    }
    // wave32 butterfly reduction
#pragma unroll
    for (int off = 16; off > 0; off >>= 1)
        acc += __shfl_xor(acc, off, 32);

    if (lane == 0) sums[row] = acc;
}

// ---------------------------------------------------------------------------
// Pass 2: out[b,o] = (x_sum[b] + w_sum[o]) * bias[o], via V_WMMA_F32_16X16X4_F32.
// One wave per 16x16 output tile.
//   A (16x4 f32): lanes 0-15 hold M=lane {K0,K1}; lanes 16-31 hold {K2,K3}.
//     -> A[m,0] = x_sum[b0+m], all other K = 0.
//   B (4x16)    : all ones (layout-invariant).
//   C (16x16)   : C[m,n] = w_sum[o0+n]  (each VGPR: N = lane&15).
//   D[m,n] = x_sum[b0+m] + w_sum[o0+n];  store D * bias[o0+n].
// D layout: VGPR r -> row b0+r (lanes 0-15) / b0+8+r (lanes 16-31), col o0+(lane&15).
// ---------------------------------------------------------------------------
__global__ void __launch_bounds__(256)
outer_bias_wmma_kernel(const float* __restrict__ sums,
                       const float* __restrict__ bias,
                       float* __restrict__ out) {
    const int wave = threadIdx.x >> 5;
    const int lane = threadIdx.x & 31;
    const int l15  = lane & 15;

    const int tile = blockIdx.x * 8 + wave;         // 0..65535
    const int tcol = tile & (O_COLS / 16 - 1);      // 128 col-tiles
    const int trow = tile >> 7;                     // 512 row-tiles
    const int b0   = trow * 16;
    const int o0   = tcol * 16;

    const float xsv = sums[b0 + l15];               // x_sum tile (L2-resident)
    const float wsv = sums[B_ROWS + o0 + l15];      // w_sum tile
    const float bv  = bias[o0 + l15];

    v2f a;
    a.x = (lane < 16) ? xsv : 0.0f;                 // K=0 column of A
    a.y = 0.0f;                                     // K=1 (and K=2/3 via upper lanes)

    v2f b;
    b.x = 1.0f;
    b.y = 1.0f;                                     // B = ones, any layout

    v8f c;
#pragma unroll
    for (int r = 0; r < 8; ++r) c[r] = wsv;         // C[m,n] = w_sum[o0+n]

    // D = A x B + C  ->  x_sum[b] + w_sum[o]
    v8f d = __builtin_amdgcn_wmma_f32_16x16x4_f32(
        /*neg_a=*/false, a, /*neg_b=*/false, b,
        /*c_mod=*/(short)0, c, /*reuse_a=*/false, /*reuse_b=*/false);

    const int rbase = b0 + ((lane >> 4) & 1) * 8;
    float* __restrict__ p = out + (size_t)rbase * O_COLS + o0 + l15;
#pragma unroll
    for (int r = 0; r < 8; ++r)
        p[(size_t)r * O_COLS] = d[r] * bv;          // 16-lane halves: 64B contiguous
}

// ---------------------------------------------------------------------------
extern "C" void kernel_launch(void* const* d_in, const int* in_sizes, int n_in,
                              void* d_out, int out_size, void* d_ws, size_t ws_size,
                              hipStream_t stream) {
    const float* x    = (const float*)d_in[0];      // [8192, 2048]
    const float* w    = (const float*)d_in[1];      // [2048, 2048]
    const float* bias = (const float*)d_in[2];      // [2048]
    float* out  = (float*)d_out;                    // [8192, 2048]
    float* sums = (float*)d_ws;                     // 10240 floats = 40 KB

    // Pass 1: 10240 rows, 8 rows (waves) per 256-thread block.
    rowsum_kernel<<<(B_ROWS + O_COLS) / 8, 256, 0, stream>>>(x, w, sums);

    // Pass 2: 512*128 = 65536 tiles, 8 tiles (waves) per block.
    outer_bias_wmma_kernel<<<65536 / 8, 256, 0, stream>>>(sums, bias, out);
}